// QuestionAnswerModel_11063835754873
// MI455X (gfx1250) — compile-verified
//
#include <hip/hip_runtime.h>
#include <math.h>

// ---------------------------------------------------------------------------
// QuestionAnswerModel (RotatE-style KGQA scoring) for MI455X / gfx1250
//   B=16, S=64, H=768, NREL=18, D=256, E=43234 (derived from in_sizes)
// ---------------------------------------------------------------------------

typedef __attribute__((ext_vector_type(2))) float v2f;
typedef __attribute__((ext_vector_type(8))) float v8f;

#define QA_B     16
#define QA_S     64
#define QA_H     768
#define QA_NREL  18
#define QA_D     256
#define QA_MARGIN 6.0f
// phase scale = PI / REL_RANGE, REL_RANGE = (6+2)/256 = 0.03125
#define QA_PHASE_SCALE 100.53096491487338f

// ---------------------------------------------------------------------------
// Kernel 1: mean-pool question_hidden (B,S,H) -> q (B,H)
// ---------------------------------------------------------------------------
__global__ __launch_bounds__(256) void qa_meanpool(const float* __restrict__ qh,
                                                   float* __restrict__ q) {
  int idx = blockIdx.x * 256 + threadIdx.x;            // 0 .. B*H-1 (12288)
  if (idx >= QA_B * QA_H) return;
  int b = idx / QA_H, h = idx - b * QA_H;
  const float* p = qh + (size_t)b * QA_S * QA_H + h;
  float s = 0.f;
#pragma unroll 8
  for (int si = 0; si < QA_S; ++si) s += p[(size_t)si * QA_H];
  q[idx] = s * (1.0f / (float)QA_S);
}

// ---------------------------------------------------------------------------
// Kernel 2: rel_scores = q @ W_rel^T + b  (WMMA f32 16x16x4, M=16 batch tile,
// two N tiles covering 18 cols padded to 32, K=768 in chunks of 4), then
// sigmoid -> relation mixture -> phase -> rotate head embedding.
// Writes rot_re[16][256] then rot_im[16][256] contiguously into ws.
// Tile-1 columns >= 18 are handled branch-free: clamped row index + 0-mask,
// keeping the WMMA K-loop straight-line (no exec-mask save/restore).
// ---------------------------------------------------------------------------
__global__ __launch_bounds__(256) void qa_relrot(const float* __restrict__ q,
                                                 const float* __restrict__ W_rel,
                                                 const float* __restrict__ b_rel,
                                                 const float* __restrict__ rel_emb,
                                                 const float* __restrict__ ent,
                                                 const int*   __restrict__ head_id,
                                                 float* __restrict__ rot) {
  __shared__ float srel[QA_B * QA_NREL];   // sigmoid(rel_scores)
  const int tid = threadIdx.x;

  if (tid < 32) {                       // wave 0, EXEC all-ones: WMMA GEMM
    const int lane = tid;
    const int row  = lane & 15;          // A-matrix M (lanes 0-15 / 16-31)
    const int kk   = (lane >> 4) << 1;   // K sub-offset: 0 or 2
    const int col  = lane & 15;          // B-matrix N within tile
    const int col1 = 16 + (col & 1);     // clamped valid row for tile 1
    const float msk = (col < QA_NREL - 16) ? 1.0f : 0.0f;

    v8f c0 = {0.f, 0.f, 0.f, 0.f, 0.f, 0.f, 0.f, 0.f};
    v8f c1 = {0.f, 0.f, 0.f, 0.f, 0.f, 0.f, 0.f, 0.f};

    const float* qrow = q + row * QA_H + kk;
    const float* w0   = W_rel + col  * QA_H + kk;   // n = col
    const float* w1   = W_rel + col1 * QA_H + kk;   // n = 16+col (masked)

    for (int k0 = 0; k0 < QA_H; k0 += 4) {
      v2f a;  a.x  = qrow[k0];      a.y  = qrow[k0 + 1];
      v2f b0; b0.x = w0[k0];        b0.y = w0[k0 + 1];
      v2f b1; b1.x = w1[k0] * msk;  b1.y = w1[k0 + 1] * msk;
      // D = A(16x4) x B(4x16) + C   (f32 in / f32 accumulate)
      c0 = __builtin_amdgcn_wmma_f32_16x16x4_f32(false, a, false, b0,
                                                 (short)0, c0, false, false);
      c1 = __builtin_amdgcn_wmma_f32_16x16x4_f32(false, a, false, b1,
                                                 (short)0, c1, false, false);
    }

    // C/D layout: VGPR v holds M = v + (lane<16 ? 0 : 8), N = lane&15
    const int mhi = (lane >> 4) << 3;
#pragma unroll
    for (int v = 0; v < 8; ++v) {
      int m = mhi + v;
      float s0 = c0[v] + b_rel[col];
      srel[m * QA_NREL + col] = 1.0f / (1.0f + __expf(-s0));
      if (col < QA_NREL - 16) {
        float s1 = c1[v] + b_rel[16 + col];
        srel[m * QA_NREL + 16 + col] = 1.0f / (1.0f + __expf(-s1));
      }
    }
  }
  __syncthreads();

  // predict_relation -> phase -> (cos,sin) -> rotate head (all 256 threads)
  for (int idx = tid; idx < QA_B * QA_D; idx += 256) {
    int m = idx >> 8, d = idx & (QA_D - 1);
    float p = 0.f;
#pragma unroll
    for (int n = 0; n < QA_NREL; ++n)
      p = fmaf(srel[m * QA_NREL + n], rel_emb[n * QA_D + d], p);
    float phase = p * QA_PHASE_SCALE;
    float cr = cosf(phase), sr = sinf(phase);
    size_t hb = (size_t)head_id[m] * (2 * QA_D);
    float reh = ent[hb + d];
    float imh = ent[hb + QA_D + d];
    rot[idx]               = reh * cr - imh * sr;   // rot_re
    rot[QA_B * QA_D + idx] = reh * sr + imh * cr;   // rot_im
  }
}

// ---------------------------------------------------------------------------
// Kernel 3: the hot loop.  One wave per entity; rot (32 KB) staged in LDS;
// each lane owns 8 dims (two float4 halves), one v_sqrt_f32 per element,
// dual accumulators for trans-op ILP, butterfly shfl_xor reduction, and the
// 16 per-entity results gathered into lanes 0..15 for a single 16-lane store.
// ---------------------------------------------------------------------------
__device__ __forceinline__ float qa_dterm(float rr, float rt, float ri, float it) {
  float a = rr - rt;
  float b = ri - it;
  return __builtin_amdgcn_sqrtf(fmaf(a, a, b * b));
}

__global__ __launch_bounds__(256) void qa_dist(const float* __restrict__ ent,
                                               const float* __restrict__ rot,
                                               float* __restrict__ out, int E) {
  __shared__ float s_rot[2 * QA_B * QA_D];   // 8192 floats = 32 KB
  for (int i = threadIdx.x; i < 2 * QA_B * QA_D; i += 256) s_rot[i] = rot[i];
  __syncthreads();
  const float4* s4 = (const float4*)s_rot;   // [0..1023]=re, [1024..2047]=im

  const int wave = threadIdx.x >> 5;
  const int lane = threadIdx.x & 31;
  const int stride = gridDim.x * 8;

  for (int e = blockIdx.x * 8 + wave; e < E; e += stride) {
    const float4* r4 = (const float4*)(ent + (size_t)e * (2 * QA_D));
    float4 reA = r4[lane];        // d = 4*lane .. 4*lane+3
    float4 reB = r4[lane + 32];   // d = 128 + 4*lane ..
    float4 imA = r4[lane + 64];
    float4 imB = r4[lane + 96];

    int en = e + stride;          // prefetch next entity row (64 B per lane)
    if (en < E)
      __builtin_prefetch(ent + (size_t)en * (2 * QA_D) + lane * 16, 0, 1);

    float res = 0.f;              // lane b collects batch-b result
#pragma unroll 4
    for (int b = 0; b < QA_B; ++b) {
      float4 ra = s4[b * 64 + lane];
      float4 rb = s4[b * 64 + 32 + lane];
      float4 ia = s4[1024 + b * 64 + lane];
      float4 ib = s4[1024 + b * 64 + 32 + lane];
      float s0, s1;
      s0  = qa_dterm(ra.x, reA.x, ia.x, imA.x);
      s1  = qa_dterm(ra.y, reA.y, ia.y, imA.y);
      s0 += qa_dterm(ra.z, reA.z, ia.z, imA.z);
      s1 += qa_dterm(ra.w, reA.w, ia.w, imA.w);
      s0 += qa_dterm(rb.x, reB.x, ib.x, imB.x);
      s1 += qa_dterm(rb.y, reB.y, ib.y, imB.y);
      s0 += qa_dterm(rb.z, reB.z, ib.z, imB.z);
      s1 += qa_dterm(rb.w, reB.w, ib.w, imB.w);
      float s = s0 + s1;
#pragma unroll
      for (int off = 16; off; off >>= 1) s += __shfl_xor(s, off, 32);
      if (lane == b) res = QA_MARGIN - s;     // v_cndmask, no store yet
    }
    if (lane < QA_B) out[(size_t)lane * E + e] = res;  // one 16-lane store
  }
}

// ---------------------------------------------------------------------------
extern "C" void kernel_launch(void* const* d_in, const int* in_sizes, int n_in,
                              void* d_out, int out_size, void* d_ws, size_t ws_size,
                              hipStream_t stream) {
  const float* qh      = (const float*)d_in[0];   // (16,64,768)
  const float* W_rel   = (const float*)d_in[1];   // (18,768)
  const float* b_rel   = (const float*)d_in[2];   // (18,)
  const float* rel_emb = (const float*)d_in[3];   // (18,256)
  const float* ent     = (const float*)d_in[4];   // (E,512)
  const int*   head_id = (const int*)d_in[5];     // (16,)
  float* out = (float*)d_out;                     // (16,E)

  const int E = in_sizes[4] / (2 * QA_D);

  float* q   = (float*)d_ws;                      // 12288 floats
  float* rot = q + QA_B * QA_H;                   // 8192 floats (re | im)

  qa_meanpool<<<(QA_B * QA_H + 255) / 256, 256, 0, stream>>>(qh, q);
  qa_relrot<<<1, 256, 0, stream>>>(q, W_rel, b_rel, rel_emb, ent, head_id, rot);
  qa_dist<<<1024, 256, 0, stream>>>(ent, rot, out, E);
}